// SoftPatternClassifier_41850161332520
// MI455X (gfx1250) — compile-verified
//
#include <hip/hip_runtime.h>
#include <hip/hip_bf16.h>

// SoftPatternClassifier for MI455X (gfx1250, wave32).
// Phase 1: fp32 GEMM (16384x300x2100) via V_WMMA_F32_16X16X4_F32.
//          Each wave owns a 32x32 macro-tile (2x2 of 16x16 WMMA tiles, 4 v8f
//          accumulators) so every A/B fragment load feeds two WMMAs -> 1:1
//          load:wmma ratio, half the redundant L2 traffic of 16x16 tiling.
// Phase 2: per-(b,p) max-plus scan over L=256 (9600 independent threads) with
//          global_prefetch_b8 of the next timestep to hide L2 latency on the
//          serial chain; then LayerNorm over P, heaviside, 150->5 matvec.

typedef float v2f __attribute__((ext_vector_type(2)));
typedef float v8f __attribute__((ext_vector_type(8)));

#define B_   64
#define L_   256
#define D_   300
#define P_   150
#define M_   7
#define K_   2100     // P * ND * M = 150*2*7
#define C_   5
#define NEG_ (-1000000000.0f)

#define CT32_ 66      // ceil(K/32) column macro-tiles
#define RT32_ 512     // (B*L)/32 row macro-tiles

__global__ __launch_bounds__(256) void spc_gemm_wmma(
    const int*   __restrict__ docs,
    const float* __restrict__ emb,
    const float* __restrict__ diags,
    const float* __restrict__ bias,
    float*       __restrict__ trans)
{
    const int wave = (blockIdx.x * 256 + threadIdx.x) >> 5;   // one wave = one 32x32 macro-tile
    const int lane = threadIdx.x & 31;
    const int tr = wave / CT32_;
    const int tc = wave - tr * CT32_;
    const int row_base = tr << 5;
    const int col_base = tc << 5;

    const int sub = lane & 15;     // row (A) / col (B) within 16x16 tile
    const int hi  = lane >> 4;     // lane-half selects K pair {0,1} vs {2,3}

    // A fragments: gathered embedding rows for output rows
    const int doc0 = docs[row_base + sub];
    const int doc1 = docs[row_base + 16 + sub];
    // B fragments: diags rows for output columns (clamp loads on ragged edge)
    const int col0 = col_base + sub;
    const int col1 = col_base + 16 + sub;
    const int col0c = col0 < K_ ? col0 : (K_ - 1);
    const int col1c = col1 < K_ ? col1 : (K_ - 1);

    const float* __restrict__ arow0 = emb   + (size_t)doc0  * D_ + 2 * hi;
    const float* __restrict__ arow1 = emb   + (size_t)doc1  * D_ + 2 * hi;
    const float* __restrict__ brow0 = diags + (size_t)col0c * D_ + 2 * hi;
    const float* __restrict__ brow1 = diags + (size_t)col1c * D_ + 2 * hi;

    v8f acc00 = {}, acc01 = {}, acc10 = {}, acc11 = {};
    #pragma unroll 3
    for (int d0 = 0; d0 < D_; d0 += 4) {
        v2f a0 = *(const v2f*)(arow0 + d0);   // K = 2*hi + {0,1} at this lane
        v2f a1 = *(const v2f*)(arow1 + d0);
        v2f b0 = *(const v2f*)(brow0 + d0);
        v2f b1 = *(const v2f*)(brow1 + d0);
        acc00 = __builtin_amdgcn_wmma_f32_16x16x4_f32(false, a0, false, b0, (short)0, acc00, false, false);
        acc01 = __builtin_amdgcn_wmma_f32_16x16x4_f32(false, a0, false, b1, (short)0, acc01, false, false);
        acc10 = __builtin_amdgcn_wmma_f32_16x16x4_f32(false, a1, false, b0, (short)0, acc10, false, false);
        acc11 = __builtin_amdgcn_wmma_f32_16x16x4_f32(false, a1, false, b1, (short)0, acc11, false, false);
    }

    // Store 4 sub-tiles; C/D layout: row = i + 8*hi within tile, col = sub.
    const float bv0 = (col0 < K_) ? bias[col0] : 0.0f;
    const float bv1 = (col1 < K_) ? bias[col1] : 0.0f;

    if (col0 < K_) {
        #pragma unroll
        for (int i = 0; i < 8; ++i) {
            const int r0 = row_base + i + (hi << 3);
            trans[(size_t)r0 * K_ + col0]        = acc00[i] + bv0;
            trans[(size_t)(r0 + 16) * K_ + col0] = acc10[i] + bv0;
        }
    }
    if (col1 < K_) {
        #pragma unroll
        for (int i = 0; i < 8; ++i) {
            const int r0 = row_base + i + (hi << 3);
            trans[(size_t)r0 * K_ + col1]        = acc01[i] + bv1;
            trans[(size_t)(r0 + 16) * K_ + col1] = acc11[i] + bv1;
        }
    }
}

__global__ __launch_bounds__(160) void spc_scan(
    const float* __restrict__ trans,
    const int*   __restrict__ doc_lens,
    const float* __restrict__ epsilons,
    const float* __restrict__ ln_gamma,
    const float* __restrict__ ln_beta,
    const float* __restrict__ W,
    const float* __restrict__ bvec,
    float*       __restrict__ out)
{
    __shared__ float sdata[P_];
    __shared__ float s_mu, s_rstd;

    const int b = blockIdx.x;
    const int p = threadIdx.x;

    float sc = NEG_;
    if (p < P_) {
        const float e0 = epsilons[p * 6 + 0];
        const float e1 = epsilons[p * 6 + 1];
        const float e2 = epsilons[p * 6 + 2];
        const float e3 = epsilons[p * 6 + 3];
        const float e4 = epsilons[p * 6 + 4];
        const float e5 = epsilons[p * 6 + 5];
        const int endi = (p < 50) ? 4 : ((p < 100) ? 5 : 6);
        const int dl = doc_lens[b];

        float h0 = 0.0f, h1 = NEG_, h2 = NEG_, h3 = NEG_, h4 = NEG_, h5 = NEG_, h6 = NEG_;

        const float* __restrict__ base =
            trans + (size_t)b * L_ * K_ + (size_t)p * 14;

        for (int t = 0; t < L_; ++t) {
            const float* tp = base + (size_t)t * K_;
            // prefetch next timestep's 56B into cache while this step computes
            __builtin_prefetch(tp + K_, 0, 3);
            // 14 floats: tm0[0..6] then tm1[0..6]; 8-byte aligned (p*56 + t*8400)
            v2f q0 = *(const v2f*)(tp + 0);
            v2f q1 = *(const v2f*)(tp + 2);
            v2f q2 = *(const v2f*)(tp + 4);
            v2f q3 = *(const v2f*)(tp + 6);
            v2f q4 = *(const v2f*)(tp + 8);
            v2f q5 = *(const v2f*)(tp + 10);
            v2f q6 = *(const v2f*)(tp + 12);
            const float t00 = q0.x, t01 = q0.y, t02 = q1.x, t03 = q1.y;
            const float t04 = q2.x, t05 = q2.y, t06 = q3.x;
            const float t10 = q3.y, t11 = q4.x, t12 = q4.y;
            const float t13 = q5.x, t14 = q5.y, t15 = q6.x;   // tm1[6] unused

            // ae[m] = max(hid[m], hid[m-1] + eps[m-1]); ae[0] = hid[0]
            const float ae0 = h0;
            const float ae1 = fmaxf(h1, h0 + e0);
            const float ae2 = fmaxf(h2, h1 + e1);
            const float ae3 = fmaxf(h3, h2 + e2);
            const float ae4 = fmaxf(h4, h3 + e3);
            const float ae5 = fmaxf(h5, h4 + e4);
            const float ae6 = fmaxf(h6, h5 + e5);

            // hid[m] = max(main[m], ae[m] + tm0[m]); main[0]=0, main[m]=ae[m-1]+tm1[m-1]
            h0 = fmaxf(0.0f,       ae0 + t00);
            h1 = fmaxf(ae0 + t10,  ae1 + t01);
            h2 = fmaxf(ae1 + t11,  ae2 + t02);
            h3 = fmaxf(ae2 + t12,  ae3 + t03);
            h4 = fmaxf(ae3 + t13,  ae4 + t04);
            h5 = fmaxf(ae4 + t14,  ae5 + t05);
            h6 = fmaxf(ae5 + t15,  ae6 + t06);

            const float ev = (endi == 4) ? h4 : ((endi == 5) ? h5 : h6);
            if (t < dl) sc = fmaxf(sc, ev);
        }
        sdata[p] = sc;
    }
    __syncthreads();

    if (threadIdx.x == 0) {
        float mu = 0.0f;
        for (int i = 0; i < P_; ++i) mu += sdata[i];
        mu *= (1.0f / P_);
        float var = 0.0f;
        for (int i = 0; i < P_; ++i) { float d = sdata[i] - mu; var += d * d; }
        var *= (1.0f / P_);
        s_mu = mu;
        s_rstd = rsqrtf(var + 1e-5f);
    }
    __syncthreads();

    if (p < P_) {
        const float normed = (sc - s_mu) * s_rstd * ln_gamma[p] + ln_beta[p];
        sdata[p] = (normed > 0.0f) ? 1.0f : 0.0f;
    }
    __syncthreads();

    if (threadIdx.x < C_) {
        float acc = bvec[threadIdx.x];
        const float* wr = W + threadIdx.x * P_;
        for (int pp = 0; pp < P_; ++pp) acc += wr[pp] * sdata[pp];
        out[b * C_ + threadIdx.x] = acc;
    }
}

extern "C" void kernel_launch(void* const* d_in, const int* in_sizes, int n_in,
                              void* d_out, int out_size, void* d_ws, size_t ws_size,
                              hipStream_t stream) {
    (void)in_sizes; (void)n_in; (void)out_size; (void)ws_size;
    const int*   docs     = (const int*)  d_in[0];
    const int*   doc_lens = (const int*)  d_in[1];
    const float* emb      = (const float*)d_in[2];
    const float* diags    = (const float*)d_in[3];
    const float* bias     = (const float*)d_in[4];
    const float* epsilons = (const float*)d_in[5];
    const float* ln_gamma = (const float*)d_in[6];
    const float* ln_beta  = (const float*)d_in[7];
    const float* W        = (const float*)d_in[8];
    const float* bvec     = (const float*)d_in[9];
    float* out   = (float*)d_out;
    float* trans = (float*)d_ws;   // B*L*K floats = 137.6 MB scratch

    // 512*66 macro-tiles, one wave (32 threads) each, 8 waves per block
    const int total_waves = RT32_ * CT32_;
    const int blocks = total_waves / 8;   // 4224
    spc_gemm_wmma<<<blocks, 256, 0, stream>>>(docs, emb, diags, bias, trans);
    spc_scan<<<B_, 160, 0, stream>>>(trans, doc_lens, epsilons,
                                     ln_gamma, ln_beta, W, bvec, out);
}